// VecNodesConv_22651657519347
// MI455X (gfx1250) — compile-verified
//
#include <hip/hip_runtime.h>

typedef float v2f __attribute__((ext_vector_type(2)));
typedef float v8f __attribute__((ext_vector_type(8)));

#define INV_SQRT_2F 0.70710678118654752440f

// Problem constants from the reference: B=2, N=50000, D_IN=D_OUT=64, K3=3.
// Columns of the batched GEMM: c = (b*N + n)*3 + k, total B*N*3 = 300000
// (exactly divisible by 16, so every wave has a full 16x16 tile and EXEC is
// all-ones for WMMA).
//
// x element (i, c)  lives at  (c/3)*192 + i*3 + (c%3)
// y element (o, c)  lives at  (c/3)*192 + o*3 + (c%3)
//
// Per wave: oTile selects 16 output channels, colTile selects 16 columns.
// D(16x16) += A(16x4 of W) * B(4x16 of X), 16 K-steps, for W_node and W_edge.
__global__ __launch_bounds__(256) void vnc_gemm_kernel(
    const float* __restrict__ x,
    const float* __restrict__ Wn,
    const float* __restrict__ We,
    float* __restrict__ out,     // receives y_node * inv_sqrt2
    float* __restrict__ yedge,   // workspace: receives y_edge
    long long cols)              // B*N*3
{
    const int lane      = threadIdx.x & 31;
    const int waveInBlk = threadIdx.x >> 5;
    const int gw        = blockIdx.x * 8 + waveInBlk;
    const int colTile   = gw >> 2;   // 16-column tile
    const int oTile     = gw & 3;    // 16-row (output channel) tile; fast index
                                     // so 4 waves share the same X columns

    const int half = lane >> 4;      // 0: lanes 0-15, 1: lanes 16-31
    const int l16  = lane & 15;

    long long c = (long long)colTile * 16 + l16;         // global column
    if (c >= cols) c = cols - 1;                         // clamp (never hit here)
    const long long colNode = c / 3;                     // b*N + n
    const int       kk      = (int)(c - colNode * 3);    // 0..2
    const size_t    xbase   = (size_t)colNode * 192 + kk;
    const int       obase   = oTile * 16;
    const int       arow    = obase + l16;               // W row held by this lane

    v8f accN = {0.f, 0.f, 0.f, 0.f, 0.f, 0.f, 0.f, 0.f};
    v8f accE = {0.f, 0.f, 0.f, 0.f, 0.f, 0.f, 0.f, 0.f};

#pragma unroll
    for (int k0 = 0; k0 < 64; k0 += 4) {
        const int ka = k0 + 2 * half;   // this half-wave holds K = ka, ka+1
        v2f aN, aE, bX;
        // A-operand: 16x4 tile of W, row-major, contiguous pair -> b64 load
        aN.x = Wn[arow * 64 + ka];
        aN.y = Wn[arow * 64 + ka + 1];
        aE.x = We[arow * 64 + ka];
        aE.y = We[arow * 64 + ka + 1];
        // B-operand: 4x16 tile of X; input channel stride is 3 floats
        bX.x = x[xbase + (size_t)(ka)     * 3];
        bX.y = x[xbase + (size_t)(ka + 1) * 3];

        accN = __builtin_amdgcn_wmma_f32_16x16x4_f32(
            false, aN, false, bX, (short)0, accN, false, false);
        accE = __builtin_amdgcn_wmma_f32_16x16x4_f32(
            false, aE, false, bX, (short)0, accE, false, false);
    }

    // C/D layout: VGPR v, lanes 0-15 -> M=v, lanes 16-31 -> M=v+8
    if ((long long)colTile * 16 + l16 < cols) {
#pragma unroll
        for (int v = 0; v < 8; ++v) {
            const int    o    = obase + v + 8 * half;
            const size_t oidx = (size_t)colNode * 192 + (size_t)o * 3 + kk;
            out[oidx]   = accN[v] * INV_SQRT_2F;
            yedge[oidx] = accE[v];
        }
    }
}

// out[b, dst[e], :, :] += norm[dst[e]] * inv_sqrt2 * yedge[b, src[e], :, :]
// One thread per (edge, batch, 4-float chunk): 192 floats/batch = 48 float4.
__global__ __launch_bounds__(256) void vnc_scatter_kernel(
    const float* __restrict__ yedge,
    const int* __restrict__ src,
    const int* __restrict__ dst,
    const float* __restrict__ norm,
    float* __restrict__ out,
    int E, int N)
{
    const long long gid   = (long long)blockIdx.x * blockDim.x + threadIdx.x;
    const long long total = (long long)E * 96;   // E * 2 batches * 48 chunks
    if (gid >= total) return;

    const int e = (int)(gid / 96);
    const int j = (int)(gid - (long long)e * 96);
    const int b = j / 48;        // batch
    const int q = j - b * 48;    // float4 chunk within (b, node)

    const int   s  = src[e];
    const int   d  = dst[e];
    const float cf = norm[d] * INV_SQRT_2F;

    const float4* vp =
        (const float4*)(yedge + ((size_t)b * N + s) * 192) + q;
    const float4 v = *vp;

    float* op = out + ((size_t)b * N + d) * 192 + (size_t)q * 4;
    atomicAdd(op + 0, cf * v.x);
    atomicAdd(op + 1, cf * v.y);
    atomicAdd(op + 2, cf * v.z);
    atomicAdd(op + 3, cf * v.w);
}

extern "C" void kernel_launch(void* const* d_in, const int* in_sizes, int n_in,
                              void* d_out, int out_size, void* d_ws, size_t ws_size,
                              hipStream_t stream) {
    const float* x    = (const float*)d_in[0];   // (B, N, 64, 3)
    const float* Wn   = (const float*)d_in[1];   // (64, 64)
    const float* We   = (const float*)d_in[2];   // (64, 64)
    const float* norm = (const float*)d_in[3];   // (N,)
    const int*   src  = (const int*)d_in[4];     // (E,)
    const int*   dst  = (const int*)d_in[5];     // (E,)
    float*       out  = (float*)d_out;           // (B, N, 64, 3)
    float*       yedg = (float*)d_ws;            // (B, N, 64, 3) scratch: 76.8 MB

    const int B = 2;
    const int N = in_sizes[3];
    const int E = in_sizes[4];

    const long long cols = (long long)B * N * 3;             // 300000
    const int numColTiles = (int)((cols + 15) / 16);         // 18750
    const int wavesTotal  = numColTiles * 4;                 // 4 oTiles each
    const int gemmBlocks  = (wavesTotal + 7) / 8;            // 8 waves / block

    vnc_gemm_kernel<<<gemmBlocks, 256, 0, stream>>>(x, Wn, We, out, yedg, cols);

    const long long scatterWork = (long long)E * 96;
    const int scatterBlocks = (int)((scatterWork + 255) / 256);

    vnc_scatter_kernel<<<scatterBlocks, 256, 0, stream>>>(yedg, src, dst, norm,
                                                          out, E, N);
}